// DeChunkLayer_13331578486933
// MI455X (gfx1250) — compile-verified
//
#include <hip/hip_runtime.h>
#include <hip/hip_bf16.h>

// ---- problem constants (match reference) ----
#define L_INNER 16384
#define L_OUTER 32768
#define DDIM    2048
#define TB      16            // timesteps per WMMA block
#define NB      (L_INNER/TB)  // 1024 blocks
#define NSUP    32
#define BPS     (NB/NSUP)     // 32 blocks per super-block
#define EPSV    1e-4f
#define DTW     4             // d-tiles (16 cols each) per wave in k_final

typedef float v2f __attribute__((ext_vector_type(2)));
typedef float v8f __attribute__((ext_vector_type(8)));

// ---- workspace layout (float offsets) ----
#define A_OFF     0u
#define PW_OFF    (A_OFF + L_INNER)
#define CD_OFF    (PW_OFF + L_INNER)          // cumulative decay, per t
#define T_OFF     (CD_OFF + L_INNER)          // 1024 * 256
#define UBLK_OFF  (T_OFF + (unsigned)NB*TB*TB)
#define ZIN_OFF   (UBLK_OFF + (unsigned)NB*DDIM)
#define USUP_OFF  (ZIN_OFF + (unsigned)NB*DDIM)
#define ZSUP_OFF  (USUP_OFF + (unsigned)NSUP*DDIM)
#define ASUP_OFF  (ZSUP_OFF + (unsigned)NSUP*DDIM)
#define START_OFF (ASUP_OFF + NSUP)           // int[L_INNER+1]

// K0: per-timestep scalars: a_t (decay with reset), pw_t (= clamped p; u = pw*x)
__global__ void k_scalars(const float* __restrict__ p, const int* __restrict__ cu,
                          int ncu, float* __restrict__ A, float* __restrict__ PW) {
  int t = blockIdx.x * blockDim.x + threadIdx.x;
  if (t >= L_INNER) return;
  float pc = fminf(fmaxf(p[t], EPSV), 1.0f - EPSV);
  float dt = -log1pf(-pc);
  float decay = expf(-dt);              // faithful to reference (== 1-p analytically)
  bool start = false;
  for (int i = 0; i < ncu - 1; ++i) start |= (cu[i] == t);
  A[t]  = start ? 0.0f : decay;
  PW[t] = pc;                           // dt*p*(x/dt) == p*x
}

// K1: per block, build lower-triangular transition matrix T and cumulative decays
__global__ void k_buildT(const float* __restrict__ A, float* __restrict__ T,
                         float* __restrict__ CD) {
  int blk = blockIdx.x * blockDim.x + threadIdx.x;
  if (blk >= NB) return;
  float a[TB];
  const float* ab = A + blk * TB;
  #pragma unroll
  for (int i = 0; i < TB; ++i) a[i] = ab[i];
  float* Tb = T + blk * TB * TB;
  for (int i = 0; i < TB; ++i) {
    Tb[i * TB + i] = 1.0f;
    float prod = 1.0f;
    for (int j = i - 1; j >= 0; --j) { prod *= a[j + 1]; Tb[i * TB + j] = prod; }
    for (int j = i + 1; j < TB; ++j) Tb[i * TB + j] = 0.0f;
  }
  float cd = 1.0f;
  for (int i = 0; i < TB; ++i) { cd *= a[i]; CD[blk * TB + i] = cd; }
}

// K2: block-end partial state (local scan with zin=0), one float4 pass over x
__global__ void k_ublk(const float4* __restrict__ x4, const float* __restrict__ A,
                       const float* __restrict__ PW, float4* __restrict__ UBLK4) {
  int d4  = blockIdx.x * blockDim.x + threadIdx.x;   // 512 float4 lanes
  int blk = blockIdx.y;
  int t0  = blk * TB;
  float4 z = make_float4(0.f, 0.f, 0.f, 0.f);
  #pragma unroll
  for (int j = 0; j < TB; ++j) {
    int t = t0 + j;
    float a = A[t], w = PW[t];
    float4 xv = x4[(size_t)t * (DDIM / 4) + d4];
    z.x = a * z.x + w * xv.x;
    z.y = a * z.y + w * xv.y;
    z.z = a * z.z + w * xv.z;
    z.w = a * z.w + w * xv.w;
  }
  UBLK4[(size_t)blk * (DDIM / 4) + d4] = z;
}

// K3a: super-block partials (32 blocks each)
__global__ void k_sup(const float4* __restrict__ UBLK4, const float* __restrict__ CD,
                      float4* __restrict__ USUP4, float* __restrict__ ASUP) {
  int d4 = blockIdx.x * blockDim.x + threadIdx.x;
  int s  = blockIdx.y;
  float4 z = make_float4(0.f, 0.f, 0.f, 0.f);
  float ap = 1.0f;
  #pragma unroll 4
  for (int q = 0; q < BPS; ++q) {
    int blk = s * BPS + q;
    float Ab = CD[blk * TB + (TB - 1)];
    float4 u = UBLK4[(size_t)blk * (DDIM / 4) + d4];
    z.x = Ab * z.x + u.x;  z.y = Ab * z.y + u.y;
    z.z = Ab * z.z + u.z;  z.w = Ab * z.w + u.w;
    ap *= Ab;
  }
  USUP4[(size_t)s * (DDIM / 4) + d4] = z;
  if (d4 == 0) ASUP[s] = ap;
}

// K3b: scan across the 32 super-blocks
__global__ void k_supscan(const float4* __restrict__ USUP4, const float* __restrict__ ASUP,
                          float4* __restrict__ ZSUP4) {
  int d4 = blockIdx.x * blockDim.x + threadIdx.x;
  float4 z = make_float4(0.f, 0.f, 0.f, 0.f);
  for (int s = 0; s < NSUP; ++s) {
    ZSUP4[(size_t)s * (DDIM / 4) + d4] = z;
    float As = ASUP[s];
    float4 u = USUP4[(size_t)s * (DDIM / 4) + d4];
    z.x = As * z.x + u.x;  z.y = As * z.y + u.y;
    z.z = As * z.z + u.z;  z.w = As * z.w + u.w;
  }
}

// K3c: incoming state for every 16-step block
__global__ void k_zin(const float4* __restrict__ UBLK4, const float* __restrict__ CD,
                      const float4* __restrict__ ZSUP4, float4* __restrict__ ZIN4) {
  int d4 = blockIdx.x * blockDim.x + threadIdx.x;
  int s  = blockIdx.y;
  float4 z = ZSUP4[(size_t)s * (DDIM / 4) + d4];
  #pragma unroll 4
  for (int q = 0; q < BPS; ++q) {
    int blk = s * BPS + q;
    ZIN4[(size_t)blk * (DDIM / 4) + d4] = z;
    float Ab = CD[blk * TB + (TB - 1)];
    float4 u = UBLK4[(size_t)blk * (DDIM / 4) + d4];
    z.x = Ab * z.x + u.x;  z.y = Ab * z.y + u.y;
    z.z = Ab * z.z + u.z;  z.w = Ab * z.w + u.w;
  }
}

// K5: prefix-sum of b -> output row range [START[t], START[t+1]) for each inner t
__global__ void k_start(const int* __restrict__ b, int* __restrict__ START) {
  __shared__ int part[1024];
  int tid  = threadIdx.x;
  int base = tid * (L_OUTER / 1024);   // 32 elements each
  int s = 0;
  #pragma unroll
  for (int i = 0; i < L_OUTER / 1024; ++i) s += b[base + i];
  part[tid] = s;
  __syncthreads();
  for (int off = 1; off < 1024; off <<= 1) {
    int add = (tid >= off) ? part[tid - off] : 0;
    __syncthreads();
    part[tid] += add;
    __syncthreads();
  }
  int c = (tid == 0) ? 0 : part[tid - 1];   // exclusive prefix
  for (int i = 0; i < L_OUTER / 1024; ++i) {
    int j = base + i;
    if (b[j]) {
      ++c;
      if (c - 1 > 0) START[c - 1] = j;      // t=0 handled below (idx clip)
    }
  }
  if (tid == 0) { START[0] = 0; START[L_INNER] = L_OUTER; }
}

// K4: intra-block scan as matmul via V_WMMA_F32_16X16X4_F32, gather fused into the store.
// One wave per (16-timestep block, 64-column group): 16 WMMAs, A fragments reused 4x.
__global__ void k_final(const float* __restrict__ x,  const float* __restrict__ PW,
                        const float* __restrict__ T,  const float* __restrict__ CD,
                        const float* __restrict__ ZIN, const int* __restrict__ START,
                        float* __restrict__ out) {
  int wave  = (blockIdx.x * blockDim.x + threadIdx.x) >> 5;
  int lane  = threadIdx.x & 31;
  int blk   = wave >> 5;                // 1024 t-blocks (DDIM/64 = 32 groups)
  int dgrp  = wave & 31;
  int t0 = blk * TB;
  int d0 = dgrp * (16 * DTW);
  int lo = lane & 15;
  int hi = lane >> 4;

  const float* Tb = T + blk * 256;

  // Preload A fragments (16x4 layout: lane -> M = lo; VGPR0 K = q*4+hi*2, VGPR1 K+1)
  v2f av[4];
  #pragma unroll
  for (int q = 0; q < 4; ++q) {
    int kA = q * 4 + hi * 2;
    av[q][0] = Tb[lo * 16 + kA];
    av[q][1] = Tb[lo * 16 + kA + 1];
  }
  float cd[8];
  #pragma unroll
  for (int v = 0; v < 8; ++v) cd[v] = CD[t0 + v + hi * 8];

  // C init per d-tile: acc[M,N] = cumdec[M] * zin[N]  (C layout: VGPR v -> M=v+hi*8, N=lo)
  v8f acc[DTW];
  #pragma unroll
  for (int k = 0; k < DTW; ++k) {
    float zin = ZIN[(size_t)blk * DDIM + d0 + k * 16 + lo];
    #pragma unroll
    for (int v = 0; v < 8; ++v) acc[k][v] = cd[v] * zin;
  }

  // K loop: 4 chunks of 4 timesteps; inner loop reuses the same A fragment
  #pragma unroll
  for (int q = 0; q < 4; ++q) {
    int tB = t0 + q * 4 + hi * 2;
    float pw0 = PW[tB], pw1 = PW[tB + 1];
    #pragma unroll
    for (int k = 0; k < DTW; ++k) {
      v2f bv;  // B 4x16: lane -> N = lo; VGPR0 K = q*4+hi*2, VGPR1 K+1
      bv[0] = pw0 * x[(size_t)tB       * DDIM + d0 + k * 16 + lo];
      bv[1] = pw1 * x[(size_t)(tB + 1) * DDIM + d0 + k * 16 + lo];
      acc[k] = __builtin_amdgcn_wmma_f32_16x16x4_f32(
          /*neg_a=*/false, av[q], /*neg_b=*/false, bv,
          /*c_mod=*/(short)0, acc[k], /*reuse_a=*/false, /*reuse_b=*/false);
    }
  }

  // Scatter: row M of the tile goes to every outer row mapped to inner index t.
  // One START lookup drives 4 d-tiles (256B per half-wave per output row).
  #pragma unroll
  for (int v = 0; v < 8; ++v) {
    int t  = t0 + v + hi * 8;
    int jb = START[t];
    int je = START[t + 1];
    float v0 = acc[0][v], v1 = acc[1][v], v2 = acc[2][v], v3 = acc[3][v];
    for (int j = jb; j < je; ++j) {
      size_t o = (size_t)j * DDIM + d0 + lo;
      out[o]      = v0;
      out[o + 16] = v1;
      out[o + 32] = v2;
      out[o + 48] = v3;
    }
  }
}

extern "C" void kernel_launch(void* const* d_in, const int* in_sizes, int n_in,
                              void* d_out, int out_size, void* d_ws, size_t ws_size,
                              hipStream_t stream) {
  const float* x  = (const float*)d_in[0];
  const float* p  = (const float*)d_in[1];
  const int*   b  = (const int*)d_in[2];
  const int*   cu = (const int*)d_in[3];
  int ncu = in_sizes[3];
  float* out = (float*)d_out;

  float* ws    = (float*)d_ws;
  float* A     = ws + A_OFF;
  float* PW    = ws + PW_OFF;
  float* CD    = ws + CD_OFF;
  float* T     = ws + T_OFF;
  float* UBLK  = ws + UBLK_OFF;
  float* ZIN   = ws + ZIN_OFF;
  float* USUP  = ws + USUP_OFF;
  float* ZSUP  = ws + ZSUP_OFF;
  float* ASUP  = ws + ASUP_OFF;
  int*   START = (int*)(ws + START_OFF);

  k_scalars<<<L_INNER / 256, 256, 0, stream>>>(p, cu, ncu, A, PW);
  k_buildT <<<NB / 256, 256, 0, stream>>>(A, T, CD);
  k_ublk   <<<dim3((DDIM / 4) / 256, NB), 256, 0, stream>>>((const float4*)x, A, PW, (float4*)UBLK);
  k_sup    <<<dim3((DDIM / 4) / 256, NSUP), 256, 0, stream>>>((const float4*)UBLK, CD, (float4*)USUP, ASUP);
  k_supscan<<<(DDIM / 4) / 256, 256, 0, stream>>>((const float4*)USUP, ASUP, (float4*)ZSUP);
  k_zin    <<<dim3((DDIM / 4) / 256, NSUP), 256, 0, stream>>>((const float4*)UBLK, CD, (float4*)ZSUP, (float4*)ZIN);
  k_start  <<<1, 1024, 0, stream>>>(b, START);
  // 1024 blocks * 32 d-groups = 32768 waves, 8 waves per 256-thread workgroup
  k_final  <<<(NB * 32) / 8, 256, 0, stream>>>(x, PW, T, CD, ZIN, START, out);
}